// Gauss_agg1_ts_50208167690573
// MI455X (gfx1250) — compile-verified
//
#include <hip/hip_runtime.h>

// Gaussian aggregation as normalized homogeneous Gram matrices:
//   out[b,s,np,k,pp,:,:] = (1/n) * sum_t [x;1][x;1]^T   (corner := 1)
// 3 distinct segment groups (nb_fr = 136 / 68 / 45) replicated to 6 outputs.
// Moment accumulation on the matrix pipe via V_WMMA_F32_16X16X4_F32:
// 4 joints packed block-diagonally in the 16x16 tile, 4 frames per WMMA.
// Main loop is branchless & unpredicated (remainder peeled into an epilogue).

typedef __attribute__((ext_vector_type(2))) float v2f;
typedef __attribute__((ext_vector_type(8))) float v8f;

#define T_FR 2048
#define NJ   25
#define FSTR (NJ * 3)   // floats per frame = 75

__device__ __forceinline__ v8f wmma_gram(v2f a, v8f c) {
  // Gram update: same registers feed A and B (B row-striping mirrors A's K layout)
  return __builtin_amdgcn_wmma_f32_16x16x4_f32(false, a, false, a,
                                               (short)0, c, false, false);
}

__global__ __launch_bounds__(224) void gauss_gram_wmma(const float* __restrict__ x,
                                                       float* __restrict__ out) {
  const int cell = blockIdx.x;          // 0..44  = g*15 + k
  const int b    = blockIdx.y;          // 0..255
  const int g    = cell / 15;           // segment group 0,1,2
  const int k    = cell % 15;           // gaussian slot
  const int nb   = (g == 0) ? 136 : ((g == 1) ? 68 : 45);
  const int f0   = k * nb;
  const int f1   = (k < 14) ? (f0 + nb) : T_FR;   // tail runs to full T (matches reference)
  const int n    = f1 - f0;
  const float inv_n = 1.0f / (float)n;

  const int wave  = threadIdx.x >> 5;   // 0..6
  const int lane  = threadIdx.x & 31;
  const int c     = lane & 3;           // y component 0..3 (3 == homogeneous 1)
  const int p     = (lane >> 2) & 3;    // joint-in-wave 0..3
  const int thalf = (lane >> 4) << 1;   // K-half: lanes>=16 hold K=2,3
  int j = wave * 4 + p;
  if (j >= NJ) j = NJ - 1;              // clamp padding problems (writes guarded later)

  // Clamp channel for the homogeneous lane so EVERY lane has a valid address:
  // c==3 lanes load x[...,2] (garbage) and cndmask it to 1.0f -> no masked loads.
  const int joff = j * 3 + ((c < 3) ? c : 2);
  const bool isOne = (c == 3);

  const float* __restrict__ xb = x + (size_t)b * T_FR * FSTR;
  const float* __restrict__ lp = xb + (size_t)(f0 + thalf) * FSTR + joff;

  v8f acc0 = {0.f,0.f,0.f,0.f,0.f,0.f,0.f,0.f};
  v8f acc1 = {0.f,0.f,0.f,0.f,0.f,0.f,0.f,0.f};

  const int nfull = (n / 8) * 8;        // frames covered without any predication
  int t = 0;

  // ---- main loop: 16 frames / 4 WMMAs per iteration, fully branchless ----
  for (; t + 16 <= nfull; t += 16) {
    float v0 = lp[0];                   // frame thalf+0
    float v1 = lp[FSTR];                // frame thalf+1
    float v2 = lp[4  * FSTR];           // frame thalf+4
    float v3 = lp[5  * FSTR];           // frame thalf+5
    float v4 = lp[8  * FSTR];           // frame thalf+8
    float v5 = lp[9  * FSTR];           // frame thalf+9
    float v6 = lp[12 * FSTR];           // frame thalf+12
    float v7 = lp[13 * FSTR];           // frame thalf+13
    lp += 16 * FSTR;
    v2f a0, a1, a2, a3;
    a0.x = isOne ? 1.0f : v0;  a0.y = isOne ? 1.0f : v1;
    a1.x = isOne ? 1.0f : v2;  a1.y = isOne ? 1.0f : v3;
    a2.x = isOne ? 1.0f : v4;  a2.y = isOne ? 1.0f : v5;
    a3.x = isOne ? 1.0f : v6;  a3.y = isOne ? 1.0f : v7;
    acc0 = wmma_gram(a0, acc0);
    acc1 = wmma_gram(a1, acc1);
    acc0 = wmma_gram(a2, acc0);
    acc1 = wmma_gram(a3, acc1);
  }
  // ---- 8-frame cleanup of the unpredicated region (runs 0 or 1 times) ----
  for (; t + 8 <= nfull; t += 8) {
    float v0 = lp[0];
    float v1 = lp[FSTR];
    float v2 = lp[4 * FSTR];
    float v3 = lp[5 * FSTR];
    lp += 8 * FSTR;
    v2f a0, a1;
    a0.x = isOne ? 1.0f : v0;  a0.y = isOne ? 1.0f : v1;
    a1.x = isOne ? 1.0f : v2;  a1.y = isOne ? 1.0f : v3;
    acc0 = wmma_gram(a0, acc0);
    acc1 = wmma_gram(a1, acc1);
  }
  // ---- predicated epilogue: rem = n % 8 in {0,2,4,5}; runs at most once ----
  const int rem = n - nfull;
  if (rem > 0) {                        // wave-uniform branch, EXEC stays all-ones
    // lp currently points at frame f0+nfull+thalf; rebase to f0+nfull.
    const float* lp2 = lp - (size_t)thalf * FSTR;
    const int fr0 = thalf, fr1 = thalf + 1, fr2 = thalf + 4, fr3 = thalf + 5;
    // clamp OOB frames to offset 0 (always in-bounds), then zero via cndmask
    float v0 = lp2[(fr0 < rem ? fr0 : 0) * FSTR];
    float v1 = lp2[(fr1 < rem ? fr1 : 0) * FSTR];
    float v2 = lp2[(fr2 < rem ? fr2 : 0) * FSTR];
    float v3 = lp2[(fr3 < rem ? fr3 : 0) * FSTR];
    v2f a0, a1;
    a0.x = isOne ? 1.0f : (fr0 < rem ? v0 : 0.0f);
    a0.y = isOne ? 1.0f : (fr1 < rem ? v1 : 0.0f);
    a1.x = isOne ? 1.0f : (fr2 < rem ? v2 : 0.0f);
    a1.y = isOne ? 1.0f : (fr3 < rem ? v3 : 0.0f);
    // zero-padded frames contribute nothing except the corner (overwritten below);
    // the homogeneous lane stays 1.0 for padded frames but pairs with x==0.
    acc0 = wmma_gram(a0, acc0);
    acc1 = wmma_gram(a1, acc1);
  }

  v8f acc = acc0 + acc1;

  // D[m][n]: m<8 -> vgpr m, lane n ; m>=8 -> vgpr m-8, lane n+16.
  // Diagonal block p lives at: p=0: lanes 0-3 vgpr 0-3 ; p=1: lanes 4-7 vgpr 4-7 ;
  //                            p=2: lanes 24-27 vgpr 0-3 ; p=3: lanes 28-31 vgpr 4-7.
  const bool hi  = lane >= 16;
  const int  ll  = hi ? (lane - 16) : lane;   // column n held by this lane
  const int  pn  = ll >> 2;                   // column-block index
  const bool writer = hi ? (pn >= 2) : (pn < 2);
  if (!writer) return;

  const int pw    = pn;            // problem (joint-in-wave) this lane writes
  const int cidx  = ll & 3;        // output column
  const int vbase = (pw & 1) * 4;  // vgpr base of the block's rows
  const int jw    = wave * 4 + pw;
  if (jw >= NJ) return;

  const int np_ = jw / 5, pp_ = jw % 5;
  const int s_start = (g == 0) ? 0 : ((g == 1) ? 1 : 3);
  const int s_cnt   = (g == 0) ? 1 : ((g == 1) ? 2 : 3);

  for (int si = 0; si < s_cnt; ++si) {
    const int s = s_start + si;
    // out shape (b, 6, NP=5, NS=15, P=5, 4, 4)
    size_t base = ((((size_t)b * 6 + s) * 5 + np_) * 15 + k) * 5 + pp_;
    float* o = out + base * 16 + cidx;
#pragma unroll
    for (int r = 0; r < 4; ++r) {
      const float val = (r == 3 && cidx == 3) ? 1.0f : acc[vbase + r] * inv_n;
      o[r * 4] = val;
    }
  }
}

extern "C" void kernel_launch(void* const* d_in, const int* in_sizes, int n_in,
                              void* d_out, int out_size, void* d_ws, size_t ws_size,
                              hipStream_t stream) {
  const float* x = (const float*)d_in[0];
  float* out = (float*)d_out;
  (void)in_sizes; (void)n_in; (void)out_size; (void)d_ws; (void)ws_size;
  dim3 grid(45, 256);            // (g*15+k, b) ; same-b WGs adjacent -> L2 reuse
  gauss_gram_wmma<<<grid, 224, 0, stream>>>(x, out);
}